// ExpertGather_2680059593069
// MI455X (gfx1250) — compile-verified
//
#include <hip/hip_runtime.h>

// CDNA5 / gfx1250: wave32, WMMA 16x16x4 F32 (full-precision matrix pipe)
// Per-wave 4x2 fragment grid (64M x 32N): each B frag feeds 4 WMMAs, each A frag 2
// -> 0.5 global b32 + 0.5 ds_load_b64 per WMMA; W L2 traffic halved vs round 2.

typedef __attribute__((ext_vector_type(2))) float v2f;
typedef __attribute__((ext_vector_type(4))) float v4f;
typedef __attribute__((ext_vector_type(8))) float v8f;

#define NB 8      // B (batch)
#define NT 8192   // T
#define NI 512    // I
#define NE 16     // E (experts)
#define NK 1024   // K (rows gathered per expert)
#define NJ 512    // J
#define LDA 516   // LDS row stride in floats (512+4 pad -> conflict-free ds_load_b64)

__launch_bounds__(256)
__global__ void expert_gather_gemm_f32(const float* __restrict__ x,
                                       const int*   __restrict__ Ind,
                                       const float* __restrict__ W,
                                       float*       __restrict__ out)
{
    __shared__ float sA[64 * LDA];   // 64 gathered rows x 512 f32 (+pad) = 132 KB

    // block decode: blockIdx.x in [0, B*E*(K/64)*(J/256)) = 4096
    const int blk  = blockIdx.x;
    const int nblk = blk & 1;             // which 256-wide N slab
    const int mblk = (blk >> 1) & 15;     // which 64-row tile of K=1024
    const int e    = (blk >> 5) & 15;     // expert
    const int b    = blk >> 9;            // batch

    const int tid   = threadIdx.x;
    const int row   = tid >> 4;           // 0..15
    const int col16 = tid & 15;           // 16 threads cooperate per row

    // ---- gather 64 rows of x[b] into LDS (coalesced float4 bursts, mostly L2 hits) ----
    const int ibase = (b * NE + e) * NK + (mblk << 6);
    #pragma unroll
    for (int p = 0; p < 4; ++p) {
        const int r   = row + (p << 4);
        const int idx = Ind[ibase + r];
        const float* __restrict__ xr = x + ((size_t)b * NT + (size_t)idx) * NI;
        float* sRow = sA + r * LDA;
        #pragma unroll
        for (int j = 0; j < 8; ++j) {
            const int o = (col16 << 2) + (j << 6);        // col16*4 + j*64
            *(v4f*)(sRow + o) = *(const v4f*)(xr + o);    // 16 lanes x 16B = 256B burst
        }
    }
    __syncthreads();

    // ---- per-wave 64(M) x 32(N) output tile: 4x2 grid of 16x16 accumulators ----
    const int wave  = tid >> 5;          // 0..7 (wave32)
    const int lane  = tid & 31;
    const int lhalf = lane >> 4;         // 0: lanes 0-15, 1: lanes 16-31
    const int l16   = lane & 15;

    const int n0 = (nblk << 8) + (wave << 5);   // 8 waves cover a 256-wide N slab

    // A fragments (ISA 32-bit A 16x4 layout): rows l16 + 16*mf
    const float* sAp0 = sA + l16 * LDA + (lhalf << 1);
    const float* sAp1 = sAp0 + 16 * LDA;
    const float* sAp2 = sAp0 + 32 * LDA;
    const float* sAp3 = sAp0 + 48 * LDA;

    // B fragments (ISA 32-bit B 4x16 layout): N sub-tiles n0, n0+16
    const float* __restrict__ pB =
        W + (size_t)e * NI * NJ + (size_t)(lhalf << 1) * NJ + n0 + l16;

    v8f c00 = {}, c01 = {};
    v8f c10 = {}, c11 = {};
    v8f c20 = {}, c21 = {};
    v8f c30 = {}, c31 = {};

    #pragma unroll 2
    for (int k = 0; k < NI; k += 4) {
        v2f a0 = *(const v2f*)(sAp0 + k);               // ds_load_b64, conflict-free
        v2f a1 = *(const v2f*)(sAp1 + k);
        v2f a2 = *(const v2f*)(sAp2 + k);
        v2f a3 = *(const v2f*)(sAp3 + k);

        const float* bk = pB + (size_t)k * NJ;
        v2f b0, b1;                                     // coalesced global_load_b32 (L2)
        b0.x = bk[0];    b0.y = bk[NJ];
        b1.x = bk[16];   b1.y = bk[NJ + 16];

        c00 = __builtin_amdgcn_wmma_f32_16x16x4_f32(false, a0, false, b0, (short)0, c00, false, false);
        c10 = __builtin_amdgcn_wmma_f32_16x16x4_f32(false, a1, false, b0, (short)0, c10, false, false);
        c20 = __builtin_amdgcn_wmma_f32_16x16x4_f32(false, a2, false, b0, (short)0, c20, false, false);
        c30 = __builtin_amdgcn_wmma_f32_16x16x4_f32(false, a3, false, b0, (short)0, c30, false, false);
        c01 = __builtin_amdgcn_wmma_f32_16x16x4_f32(false, a0, false, b1, (short)0, c01, false, false);
        c11 = __builtin_amdgcn_wmma_f32_16x16x4_f32(false, a1, false, b1, (short)0, c11, false, false);
        c21 = __builtin_amdgcn_wmma_f32_16x16x4_f32(false, a2, false, b1, (short)0, c21, false, false);
        c31 = __builtin_amdgcn_wmma_f32_16x16x4_f32(false, a3, false, b1, (short)0, c31, false, false);
    }

    // ---- store 64x32 tile; C layout: VGPR r -> row M = r + 8*lhalf (+16*mf), N = lane%16 ----
    float* __restrict__ po =
        out + ((((size_t)b * NE + e) * NK + (mblk << 6) + (lhalf << 3)) * NJ) + n0 + l16;
    #pragma unroll
    for (int r = 0; r < 8; ++r) {
        po[(size_t)r * NJ + 0]          = c00[r];
        po[(size_t)r * NJ + 16]         = c01[r];
        po[(size_t)(r + 16) * NJ + 0]   = c10[r];
        po[(size_t)(r + 16) * NJ + 16]  = c11[r];
        po[(size_t)(r + 32) * NJ + 0]   = c20[r];
        po[(size_t)(r + 32) * NJ + 16]  = c21[r];
        po[(size_t)(r + 48) * NJ + 0]   = c30[r];
        po[(size_t)(r + 48) * NJ + 16]  = c31[r];
    }
}

extern "C" void kernel_launch(void* const* d_in, const int* in_sizes, int n_in,
                              void* d_out, int out_size, void* d_ws, size_t ws_size,
                              hipStream_t stream) {
    (void)in_sizes; (void)n_in; (void)out_size; (void)d_ws; (void)ws_size;
    const float* x   = (const float*)d_in[0];   // [B,T,I] f32
    const int*   Ind = (const int*)  d_in[1];   // [B,E,K] i32
    const float* W   = (const float*)d_in[2];   // [E,I,J] f32
    float*       out = (float*)d_out;           // [B,E,K,J] f32

    const int grid = NB * NE * (NK / 64) * (NJ / 256);  // 4096 workgroups
    expert_gather_gemm_f32<<<grid, 256, 0, stream>>>(x, Ind, W, out);
}